// Membership_norm_cuda_43525198577708
// MI455X (gfx1250) — compile-verified
//
#include <hip/hip_runtime.h>
#include <hip/hip_bf16.h>
#include <math.h>

typedef __attribute__((ext_vector_type(2))) float v2f;
typedef __attribute__((ext_vector_type(4))) float v4f;
typedef __attribute__((ext_vector_type(8))) float v8f;

#define NB 8
#define DD 64
#define WH 16384
#define CC 80

// dist^T tile GEMM: D(16c x 16p) += A(16c x 4d) * B(4d x 16p)
//   A = lambda^T / (-2*lambda*c)^T fragments, pre-swizzled in LDS so each
//       lane's {L0,L1,G0,G1} quad is one aligned ds_load_b128
//   B = x tiles (contiguous along p -> coalesced), squared in VALU for term 1
__global__ __launch_bounds__(256)
void membership_norm_kernel(const float* __restrict__ x,
                            const float* __restrict__ cen,
                            const float* __restrict__ lam,
                            float* __restrict__ out) {
    // sF[(kc*2 + half)*CC + c] = { L[k0][c], L[k0+1][c], G[k0][c], G[k0+1][c] }
    //   where k0 = kc*4 + 2*half  (exactly the A-fragment layout of
    //   V_WMMA_F32_16X16X4_F32: lanes 0-15 hold K0/K1, lanes 16-31 hold K2/K3)
    __shared__ v4f  sF[16 * 2 * CC];   // 40 KB
    __shared__ float sK[CC];           // sum_d lambda*c*c

    const int tid = threadIdx.x;

    // ---- stage operands into LDS in fragment order ----
    float* sFf = (float*)sF;
    for (int i = tid; i < DD * CC; i += 256) {
        const int d = i / CC;
        const int c = i - d * CC;
        const float l  = lam[i];
        const float cv = cen[i];
        const int slot = (((d >> 2) << 1) + ((d >> 1) & 1)) * CC + c;
        const int j = d & 1;
        sFf[slot * 4 + j]     = l;                 // L half of the quad
        sFf[slot * 4 + 2 + j] = -2.0f * l * cv;    // G half of the quad
    }
    if (tid < CC) {
        float s = 0.0f;
#pragma unroll 8
        for (int d = 0; d < DD; ++d) {
            const float cv = cen[d * CC + tid];
            s = fmaf(lam[d * CC + tid] * cv, cv, s);
        }
        sK[tid] = s;
    }
    __syncthreads();

    const int wave = tid >> 5;
    const int lane = tid & 31;
    const int half = lane >> 4;   // 0: lanes 0-15, 1: lanes 16-31
    const int lm   = lane & 15;

    // 8192 strips of 16 consecutive p; 1024 strips per image n (wh=16384)
    const int strip = blockIdx.x * 8 + wave;
    const int n     = strip >> 10;
    const int pbase = (strip & 1023) << 4;

    const float* xb = x + (size_t)n * DD * WH + pbase;   // x[n][d][pbase + *]

    v8f acc[5] = {};   // 5 c-tiles x (16x16 f32 accumulator = 8 VGPRs)

#pragma unroll 4
    for (int kc = 0; kc < 16; ++kc) {
        // B fragment: x[n][k0..k0+1][pbase+lm]
        const int k0 = kc * 4 + 2 * half;
        const float b0 = xb[(size_t)k0 * WH + lm];
        const float b1 = xb[(size_t)(k0 + 1) * WH + lm];
        v2f bx  = { b0, b1 };
        v2f bx2 = { b0 * b0, b1 * b1 };

        const v4f* fr = sF + ((kc << 1) + half) * CC + lm;
#pragma unroll
        for (int t = 0; t < 5; ++t) {
            const v4f f = fr[t * 16];                       // one ds_load_b128
            v2f aL = __builtin_shufflevector(f, f, 0, 1);   // adjacent VGPR pair
            v2f aG = __builtin_shufflevector(f, f, 2, 3);   // adjacent VGPR pair
            // dist += lambda^T * x^2
            acc[t] = __builtin_amdgcn_wmma_f32_16x16x4_f32(
                false, aL, false, bx2, (short)0, acc[t], false, false);
            // dist += (-2*lambda*c)^T * x
            acc[t] = __builtin_amdgcn_wmma_f32_16x16x4_f32(
                false, aG, false, bx, (short)0, acc[t], false, false);
        }
    }

    // ---- fused epilogue: + const, exp, clamp, coalesced store along p ----
    float* ob = out + (size_t)n * CC * WH + pbase + lm;
#pragma unroll
    for (int t = 0; t < 5; ++t) {
#pragma unroll
        for (int v = 0; v < 8; ++v) {
            const int c = t * 16 + v + 8 * half;   // C/D layout: M = v + 8*half
            float dist = acc[t][v] + sK[c];
            float val  = __expf(-dist);
            val = fmaxf(val, 1e-6f);
            ob[(size_t)c * WH] = val;              // 16 contiguous floats per half-wave
        }
    }
}

extern "C" void kernel_launch(void* const* d_in, const int* in_sizes, int n_in,
                              void* d_out, int out_size, void* d_ws, size_t ws_size,
                              hipStream_t stream) {
    const float* x   = (const float*)d_in[0];   // (8, 64, 16384)
    const float* cen = (const float*)d_in[1];   // (64, 80)
    const float* lam = (const float*)d_in[2];   // (64, 80)
    float* out = (float*)d_out;                 // (8, 80, 16384)

    // 8192 p-strips / 8 waves per block = 1024 blocks of 256 threads (8 wave32)
    membership_norm_kernel<<<dim3(1024), dim3(256), 0, stream>>>(x, cen, lam, out);
}